// FrequencyLoss_111669150024
// MI455X (gfx1250) — compile-verified
//
#include <hip/hip_runtime.h>

// Haar-DWT frequency loss, MI455X (gfx1250, wave32).
// Bandwidth-bound: 64 MiB streamed once -> ~2.9us floor @ 23.3 TB/s.
// WMMA f32 16x16x4 with all-ones B is used as the wave-level accumulator.

typedef float v2f __attribute__((ext_vector_type(2)));
typedef float v8f __attribute__((ext_vector_type(8)));

#define BB 32
#define HH 512
#define WW 512
#define ROWPAIRS (HH / 2)               // 256
#define QCOLS (WW / 4)                  // 128 float4s per row
#define NITEMS (BB * ROWPAIRS * QCOLS)  // 1,048,576 work items (2 Haar blocks each)
#define NBLOCKS 1024
#define NTHREADS 256
#define TOTTHREADS (NBLOCKS * NTHREADS) // 262,144
#define NITERS (NITEMS / TOTTHREADS)    // 4 (exact -> EXEC all-1s everywhere)

__global__ __launch_bounds__(NTHREADS) void haar_loss_partial(
    const float* __restrict__ in, const float* __restrict__ tg,
    float* __restrict__ ws)
{
    const float4* in4 = reinterpret_cast<const float4*>(in);
    const float4* tg4 = reinterpret_cast<const float4*>(tg);

    v8f c_low  = {};
    v8f c_high = {};
    const v2f ones = {1.0f, 1.0f};

    const unsigned tid = blockIdx.x * NTHREADS + threadIdx.x;

#pragma unroll
    for (int it = 0; it < NITERS; ++it) {
        unsigned idx   = tid + (unsigned)it * TOTTHREADS;
        unsigned col   = idx & (QCOLS - 1);
        unsigned even4 = 2u * idx - col;      // = b*65536 + j*256 + col
        unsigned odd4  = even4 + (WW / 4);    // next image row

        float4 ie = in4[even4];
        float4 io = in4[odd4];
        float4 te = tg4[even4];
        float4 to = tg4[odd4];

        // diffs: a = even/even, b = even/odd, c = odd/even, d = odd/odd
        float a0 = ie.x - te.x, b0 = ie.y - te.y, c0 = io.x - to.x, d0 = io.y - to.y;
        float a1 = ie.z - te.z, b1 = ie.w - te.w, c1 = io.z - to.z, d1 = io.w - to.w;

        // 2*Haar (0.5 factor folded into final scaling)
        float p0 = a0 + b0, q0 = c0 + d0, r0 = a0 + c0, s0 = b0 + d0, tt0 = a0 + d0, u0 = b0 + c0;
        float p1 = a1 + b1, q1 = c1 + d1, r1 = a1 + c1, s1 = b1 + d1, tt1 = a1 + d1, u1 = b1 + c1;

        v2f alow  = { fabsf(p0 + q0), fabsf(p1 + q1) };
        v2f ahigh = { fabsf(q0 - p0) + fabsf(s0 - r0) + fabsf(tt0 - u0),
                      fabsf(q1 - p1) + fabsf(s1 - r1) + fabsf(tt1 - u1) };

        // D = A(16x4) * ones(4x16) + C : every column of D accumulates the
        // row-sums of A, so column 0 summed over M = total of all 64 values.
        c_low  = __builtin_amdgcn_wmma_f32_16x16x4_f32(false, alow,  false, ones,
                                                       (short)0, c_low,  false, false);
        c_high = __builtin_amdgcn_wmma_f32_16x16x4_f32(false, ahigh, false, ones,
                                                       (short)0, c_high, false, false);
    }

    // Per-lane: sum 8 C VGPRs (rows M=0..7 of this lane's column, or M=8..15
    // for lanes 16..31). xor-16 folds the two halves -> full wave sum.
    float sl = c_low[0] + c_low[1] + c_low[2] + c_low[3]
             + c_low[4] + c_low[5] + c_low[6] + c_low[7];
    float sh = c_high[0] + c_high[1] + c_high[2] + c_high[3]
             + c_high[4] + c_high[5] + c_high[6] + c_high[7];
    sl += __shfl_xor(sl, 16, 32);
    sh += __shfl_xor(sh, 16, 32);

    __shared__ float wl[NTHREADS / 32];
    __shared__ float wh[NTHREADS / 32];
    unsigned lane = threadIdx.x & 31u;
    unsigned wave = threadIdx.x >> 5;
    if (lane == 0) { wl[wave] = sl; wh[wave] = sh; }
    __syncthreads();
    if (threadIdx.x == 0) {
        float tl = 0.0f, th = 0.0f;
#pragma unroll
        for (int i = 0; i < NTHREADS / 32; ++i) { tl += wl[i]; th += wh[i]; }
        ws[2u * blockIdx.x + 0] = tl;
        ws[2u * blockIdx.x + 1] = th;
    }
}

__global__ __launch_bounds__(32) void haar_loss_final(
    const float* __restrict__ ws, float* __restrict__ out)
{
    // One wave; fixed lane->index mapping + fixed shuffle tree => deterministic.
    float tl = 0.0f, th = 0.0f;
    for (int i = threadIdx.x; i < NBLOCKS; i += 32) {
        tl += ws[2 * i + 0];
        th += ws[2 * i + 1];
    }
#pragma unroll
    for (int off = 16; off > 0; off >>= 1) {
        tl += __shfl_xor(tl, off, 32);
        th += __shfl_xor(th, off, 32);
    }
    if (threadIdx.x == 0) {
        const float nlow = (float)(BB * (HH / 2) * (WW / 2)); // 2,097,152
        out[0] = 0.5f * tl / nlow;                 // mean |ll_i - ll_t|
        out[1] = 0.5f * th / (3.0f * nlow);        // mean(hf_abs)/3
    }
}

extern "C" void kernel_launch(void* const* d_in, const int* in_sizes, int n_in,
                              void* d_out, int out_size, void* d_ws, size_t ws_size,
                              hipStream_t stream) {
    const float* in = (const float*)d_in[0];
    const float* tg = (const float*)d_in[1];
    float* ws  = (float*)d_ws;   // needs 2*NBLOCKS floats = 8 KB
    float* out = (float*)d_out;  // 2 fp32 scalars

    haar_loss_partial<<<NBLOCKS, NTHREADS, 0, stream>>>(in, tg, ws);
    haar_loss_final<<<1, 32, 0, stream>>>(ws, out);
}